// OutputHead_71554155151995
// MI455X (gfx1250) — compile-verified
//
#include <hip/hip_runtime.h>
#include <cstdint>

// Problem constants (match the reference)
#define R_CONST 196000
#define PAD 14
#define MUL 64
#define FEAT 320
#define SD 128

typedef __attribute__((ext_vector_type(2))) float v2f;
typedef __attribute__((ext_vector_type(8))) float v8f;
typedef unsigned int u32x4 __attribute__((ext_vector_type(4)));
typedef int i32x4 __attribute__((ext_vector_type(4)));
typedef int i32x8 __attribute__((ext_vector_type(8)));

// ---------------------------------------------------------------------------
// Kernel A: per-residue channel mix via V_WMMA_F32_16X16X4_F32, with the
// feature tile staged into LDS by the Tensor Data Mover (TDM).
//   Tile per block: features[r0 .. r0+127, 128:320]  (2D strided tile)
//   TDM trick: tensor_dim0=192 (valid cols from tile start) but tile_dim0=196
//   -> 4 OOB-zero floats appended per row = bank-conflict-free 196-f32 LDS
//   row stride. tensor_dim1 = R - r0 zero-fills tail rows past the end.
//   D(p=16, n=16 residues) = Wt(16x64) * X(64x16), K split into 16 chunks of 4.
// ---------------------------------------------------------------------------
__global__ __launch_bounds__(256) void mix_wmma_kernel(
    const float* __restrict__ features,  // [R,320]
    const float* __restrict__ w_base,    // [64,1]
    const float* __restrict__ w_rel,     // [64,14]
    float* __restrict__ out_base,        // [R,3]   (head of d_out)
    float* __restrict__ ws_rel)          // [R,14,3] workspace
{
    extern __shared__ float lds[];          // [128][196] padded rows
    const int ROWSTRIDE = 196;
    const int r0 = blockIdx.x * 128;

    // --- TDM: DMA the strided feature tile into LDS (wave 0 issues once) ---
    if (threadIdx.x < 32) {
        uint64_t gaddr = (uint64_t)(const void*)(features + (size_t)r0 * FEAT + SD);
        uint32_t ldsoff = (uint32_t)(uintptr_t)&lds[0];

        // D# group 0: count=1 | lds_addr | global_addr[56:0] | type=2
        u32x4 g0;
        g0[0] = 1u;                                   // count=1, user descriptor
        g0[1] = ldsoff;                               // LDS byte address
        g0[2] = (uint32_t)gaddr;                      // global_addr[31:0]
        g0[3] = (uint32_t)((gaddr >> 32) & 0x01FFFFFFu) | 0x80000000u; // [56:32] | type=2

        // D# group 1: data_size=4B, dims/strides (element units)
        uint32_t td1 = (uint32_t)(R_CONST - r0);      // rows remaining from tile start
        i32x8 g1;
        g1[0] = (int)(2u << 16);                      // wg_mask=0, data_size=4B, no flags
        g1[1] = (int)((192u & 0xFFFFu) << 16);        // barrier_addr=0 | tensor_dim0.lo
        g1[2] = (int)((192u >> 16) | ((td1 & 0xFFFFu) << 16)); // dim0.hi | dim1.lo
        g1[3] = (int)((td1 >> 16) | (196u << 16));    // dim1.hi | tile_dim0=196
        g1[4] = (int)128u;                            // tile_dim1=128 | tile_dim2=0
        g1[5] = (int)320u;                            // tensor_dim0_stride[31:0]
        g1[6] = 0;                                    // dim0_stride.hi | dim1_stride.lo
        g1[7] = 0;                                    // dim1_stride.hi

        i32x4 gz = {0, 0, 0, 0};
#if defined(__clang_major__) && __clang_major__ >= 23
        i32x8 gz8 = {0, 0, 0, 0, 0, 0, 0, 0};
        __builtin_amdgcn_tensor_load_to_lds(g0, g1, gz, gz, gz8, 0);
#else
        __builtin_amdgcn_tensor_load_to_lds(g0, g1, gz, gz, 0);
#endif
        __builtin_amdgcn_s_wait_tensorcnt(0);
    }

    // --- Build A fragments: A[p,k] = W[k,p] * NORM (constant per wave) ---
    const int lane = threadIdx.x & 31;
    const int wave = threadIdx.x >> 5;
    const int pcol = lane & 15;             // output column p this lane carries
    const int koff = (lane >> 4) << 1;      // lanes 16-31 carry K = kb+2, kb+3
    const float norm = 0.125f;              // 1/sqrt(64)
    v2f afrag[16];
#pragma unroll
    for (int kb = 0; kb < 16; ++kb) {
        int k0 = kb * 4 + koff;
        float a0, a1;
        if (pcol == 0) {
            a0 = w_base[k0];
            a1 = w_base[k0 + 1];
        } else if (pcol <= PAD) {
            a0 = w_rel[k0 * PAD + (pcol - 1)];
            a1 = w_rel[(k0 + 1) * PAD + (pcol - 1)];
        } else {                            // p == 15: zero padding column
            a0 = 0.f;
            a1 = 0.f;
        }
        afrag[kb].x = a0 * norm;
        afrag[kb].y = a1 * norm;
    }
    __syncthreads();                        // TDM data visible to all waves

    // --- WMMA chain: each wave owns 16 residues, 3 accumulator passes (x) ---
    const int n     = lane & 15;            // residue slot within the tile
    const int rbase = r0 + wave * 16;
    const float* rowp = &lds[(wave * 16 + n) * ROWSTRIDE];

#pragma unroll
    for (int x = 0; x < 3; ++x) {
        v8f acc = {0.f, 0.f, 0.f, 0.f, 0.f, 0.f, 0.f, 0.f};
#pragma unroll
        for (int kb = 0; kb < 16; ++kb) {
            int k0 = kb * 4 + koff;
            v2f b;
            b.x = rowp[k0 * 3 + x];         // vec[r, k0,   x]
            b.y = rowp[k0 * 3 + 3 + x];     // vec[r, k0+1, x]
            acc = __builtin_amdgcn_wmma_f32_16x16x4_f32(
                false, afrag[kb], false, b, (short)0, acc, false, false);
        }
        // D layout: VGPR i, lanes 0-15 -> p = i, lanes 16-31 -> p = i + 8
        int r = rbase + n;
        if (r < R_CONST) {
            int pbase = (lane >= 16) ? 8 : 0;
#pragma unroll
            for (int i = 0; i < 8; ++i) {
                int p   = pbase + i;
                float v = acc[i];
                if (p == 0)
                    out_base[(size_t)r * 3 + x] = v;
                else if (p <= PAD)
                    ws_rel[((size_t)r * PAD + (p - 1)) * 3 + x] = v;
                // p == 15 is padding: discard
            }
        }
    }
}

// ---------------------------------------------------------------------------
// Kernel B: per-residue start offsets from the sorted residue index.
// ---------------------------------------------------------------------------
__global__ __launch_bounds__(256) void offsets_kernel(
    const int* __restrict__ idx, int* __restrict__ offs, int N)
{
    int j = blockIdx.x * blockDim.x + threadIdx.x;
    if (j >= N) return;
    int r = idx[j];
    if (j == 0 || idx[j - 1] != r) offs[r] = j;   // every residue has >=1 atom
}

// ---------------------------------------------------------------------------
// Kernel C: ragged gather  unpadded[j] = rel[idx[j], j - offs[idx[j]]]
//           (rel array is L2-resident: 33 MB << 192 MB)
// ---------------------------------------------------------------------------
__global__ __launch_bounds__(256) void scatter_kernel(
    const int* __restrict__ idx, const int* __restrict__ offs,
    const float* __restrict__ ws_rel, float* __restrict__ out_un, int N)
{
    int j = blockIdx.x * blockDim.x + threadIdx.x;
    if (j >= N) return;
    int r   = idx[j];
    int pos = j - offs[r];
    const float* src = ws_rel + ((size_t)r * PAD + pos) * 3;
    float* dst = out_un + (size_t)j * 3;
    dst[0] = src[0];
    dst[1] = src[1];
    dst[2] = src[2];
}

extern "C" void kernel_launch(void* const* d_in, const int* in_sizes, int n_in,
                              void* d_out, int out_size, void* d_ws, size_t ws_size,
                              hipStream_t stream)
{
    const float* features = (const float*)d_in[0];
    const float* w_base   = (const float*)d_in[1];
    const float* w_rel    = (const float*)d_in[2];
    const int*   resIdx   = (const int*)d_in[3];
    const int N = in_sizes[3];

    float* out_base = (float*)d_out;
    float* out_un   = (float*)d_out + (size_t)R_CONST * 3;

    float* ws_rel = (float*)d_ws;                                   // R*14*3 f32
    int*   offs   = (int*)((char*)d_ws +
                           (size_t)R_CONST * PAD * 3 * sizeof(float));

    const int gridA = (R_CONST + 127) / 128;
    const size_t ldsBytes = 128 * 196 * sizeof(float);
    mix_wmma_kernel<<<gridA, 256, ldsBytes, stream>>>(
        features, w_base, w_rel, out_base, ws_rel);

    const int gridN = (N + 255) / 256;
    offsets_kernel<<<gridN, 256, 0, stream>>>(resIdx, offs, N);
    scatter_kernel<<<gridN, 256, 0, stream>>>(resIdx, offs, ws_rel, out_un, N);
}